// DispersionD3_24077586661880
// MI455X (gfx1250) — compile-verified
//
#include <hip/hip_runtime.h>
#include <cstdint>

// ---------------------------------------------------------------------------
// D3 dispersion correction, MI455X (gfx1250, wave32).
// Phase 1: coordination numbers via logistic counting + global f32 atomics.
// Phase 2: per-pair 5x5 Gaussian C6 interpolation. The three 25-float table
//          rows per pair are fetched with GLOBAL_LOAD_ASYNC_TO_LDS (ASYNCcnt),
//          double-buffered in LDS and software-pipelined: iteration i+1's
//          21 async gathers are in flight while iteration i computes.
// ---------------------------------------------------------------------------

#define ANG2BOHR 1.8897261258369282f
#define K1C 16.0f
#define K2C (4.0f / 3.0f)
#define NELEM 95

// -K3 * log2(e) : v_exp_f32 is a base-2 exponential
#define NEG_K3_LOG2E (-5.770780163555852f)
#define NEG_K1_LOG2E (-23.083120654223408f)   // -K1 * log2(e)

constexpr int TPB = 64;         // threads per block for the energy kernel
constexpr int STRIDE_DW = 84;   // 3 tables * 28 dwords per buffer (16B aligned)

__device__ __forceinline__ float fast_exp2(float x) {
#if __has_builtin(__builtin_amdgcn_exp2f)
    return __builtin_amdgcn_exp2f(x);
#else
    return __expf(x * 0.6931471805599453f);
#endif
}

// --- CDNA5 async global->LDS helpers (inline asm; tracked by ASYNCcnt) -----
__device__ __forceinline__ void async_b128(uint32_t lds, const float* g) {
    asm volatile("global_load_async_to_lds_b128 %0, %1, off"
                 :: "v"(lds), "v"(g) : "memory");
}
__device__ __forceinline__ void async_b32(uint32_t lds, const float* g) {
    asm volatile("global_load_async_to_lds_b32 %0, %1, off"
                 :: "v"(lds), "v"(g) : "memory");
}
__device__ __forceinline__ void wait_async0() {      // all batches done
    asm volatile("s_wait_asynccnt 0x0" ::: "memory");
}
__device__ __forceinline__ void wait_async_le21() {  // previous batch done
    asm volatile("s_wait_asynccnt 0x15" ::: "memory");
}

// 21 async gathers: order6[25] @ +0B, cn_a[25] @ +112B, cn_b[25] @ +224B
__device__ __forceinline__ void issue_batch(uint32_t lds, const float* g6,
                                            const float* ga, const float* gb) {
    #pragma unroll
    for (int k = 0; k < 6; ++k) async_b128(lds + k * 16, g6 + 4 * k);
    async_b32(lds + 96, g6 + 24);
    #pragma unroll
    for (int k = 0; k < 6; ++k) async_b128(lds + 112 + k * 16, ga + 4 * k);
    async_b32(lds + 208, ga + 24);
    #pragma unroll
    for (int k = 0; k < 6; ++k) async_b128(lds + 224 + k * 16, gb + 4 * k);
    async_b32(lds + 320, gb + 24);
}

// --- kernel 0: zero workspace (coordnums[natoms] + accumulator[1]) ---------
__global__ void zero_ws_kernel(float* __restrict__ ws, int n) {
    int i = blockIdx.x * blockDim.x + threadIdx.x;
    if (i < n) ws[i] = 0.0f;
}

// --- kernel 1: coordination numbers ----------------------------------------
__global__ void cn_kernel(const int* __restrict__ species,
                          const int* __restrict__ idx12,
                          const float* __restrict__ dist,
                          const float* __restrict__ covr,
                          float* __restrict__ coord,
                          int npairs) {
    int stride = gridDim.x * blockDim.x;
    for (int p = blockIdx.x * blockDim.x + threadIdx.x; p < npairs; p += stride) {
        int ai = idx12[p];
        int aj = idx12[npairs + p];
        float d = dist[p] * ANG2BOHR;
        float cov = covr[species[ai]] + covr[species[aj]];
        float x = K2C * cov / d - 1.0f;                 // K1*(x); exp(-K1*x)
        float c = 1.0f / (1.0f + fast_exp2(NEG_K1_LOG2E * x));
        atomicAdd(&coord[ai], c);
        atomicAdd(&coord[aj], c);
    }
}

// --- kernel 2: pair dispersion energies (pipelined async gathers) ----------
__global__ void __launch_bounds__(TPB)
energy_kernel(const int* __restrict__ species,
              const int* __restrict__ idx12,
              const float* __restrict__ dist,
              const float* __restrict__ cutoff_tab,   // [95*95]
              const float* __restrict__ q_tab,        // [95*95]
              const float* __restrict__ tab_c6,       // [95*95*25]
              const float* __restrict__ tab_a,        // [95*95*25]
              const float* __restrict__ tab_b,        // [95*95*25]
              const float* __restrict__ coord,        // [natoms]
              const float* __restrict__ s6p, const float* __restrict__ s8p,
              const float* __restrict__ a1p, const float* __restrict__ a2p,
              float* __restrict__ acc,
              int npairs) {
    __shared__ float stage[TPB * 2 * STRIDE_DW];   // 43,008 B per block

    const int tid = threadIdx.x;
    float* st0 = &stage[tid * (2 * STRIDE_DW)];
    float* st1 = st0 + STRIDE_DW;
    const uint32_t lds0 = (uint32_t)(uintptr_t)st0;  // low 32 bits = LDS addr
    const uint32_t lds1 = lds0 + STRIDE_DW * 4;

    const float s6 = s6p[0], s8 = s8p[0], a1 = a1p[0], a2 = a2p[0];

    float accl = 0.0f;
    const int gsz = gridDim.x * TPB;
    const int n_iter = (npairs + gsz - 1) / gsz;
    const int base = blockIdx.x * TPB + tid;

    // ---- prologue: issue batch for iteration 0 into buffer 0 ----
    int p0 = base;
    int pc = (p0 < npairs) ? p0 : npairs - 1;        // clamp, lanes converged
    int ai = idx12[pc];
    int aj = idx12[npairs + pc];
    int ss = species[ai] * NELEM + species[aj];
    issue_batch(lds0, tab_c6 + ss * 25, tab_a + ss * 25, tab_b + ss * 25);

    for (int it = 0; it < n_iter; ++it) {
        // current pair state (fetched last iteration / prologue)
        const int   cai = ai, caj = aj, css = ss, cpc = pc;
        const bool  valid = (base + it * gsz) < npairs;
        float* stc = (it & 1) ? st1 : st0;

        // ---- issue next iteration's gathers into the other buffer ----
        const bool have_next = (it + 1) < n_iter;    // uniform branch
        if (have_next) {
            int pn = base + (it + 1) * gsz;
            pc = (pn < npairs) ? pn : npairs - 1;
            ai = idx12[pc];
            aj = idx12[npairs + pc];
            ss = species[ai] * NELEM + species[aj];
            uint32_t ldsn = (it & 1) ? lds0 : lds1;
            issue_batch(ldsn, tab_c6 + ss * 25, tab_a + ss * 25, tab_b + ss * 25);
        }

        // ---- overlap: damping terms for the current pair ----
        float d   = dist[cpc] * ANG2BOHR;
        float cni = coord[cai];
        float cnj = coord[caj];
        float cr  = cutoff_tab[css];
        float q   = q_tab[css];
        float r   = a1 * cr + a2;
        float d2 = d * d, d6 = d2 * d2 * d2, d8 = d6 * d2;
        float r2 = r * r, r6 = r2 * r2 * r2, r8 = r6 * r2;
        float inv6 = 1.0f / (d6 + r6);
        float inv8 = 1.0f / (d8 + r8);

        // previous batch landed when <=21 remain (async loads retire in order)
        if (have_next) wait_async_le21(); else wait_async0();

        // ---- 5x5 Gaussian interpolation from LDS staging ----
        float w = 0.0f, z = 0.0f;
        #pragma unroll
        for (int e = 0; e < 25; ++e) {
            float a  = stc[e];          // order6
            float ca = stc[28 + e];     // cn_a
            float cb = stc[56 + e];     // cn_b
            float dx = cni - ca;
            float dy = cnj - cb;
            float t  = fmaf(dx, dx, dy * dy);
            float g  = fast_exp2(t * NEG_K3_LOG2E);
            w += g;
            z = fmaf(a, g, z);
        }
        float c6 = z / w;
        float e2 = s6 * c6 * inv6 + 3.0f * s8 * c6 * q * inv8;
        if (valid) accl += e2;
    }

    // wave32 reduction, then one atomic per wave
    #pragma unroll
    for (int m = 16; m > 0; m >>= 1) accl += __shfl_xor(accl, m, 32);
    if ((tid & 31) == 0) atomicAdd(acc, accl);
}

// --- kernel 3: finalize -----------------------------------------------------
__global__ void finalize_kernel(const float* __restrict__ energies,
                                const float* __restrict__ acc,
                                float* __restrict__ out) {
    if (blockIdx.x == 0 && threadIdx.x == 0) {
        out[0] = energies[0] - acc[0];
    }
}

// ---------------------------------------------------------------------------
extern "C" void kernel_launch(void* const* d_in, const int* in_sizes, int n_in,
                              void* d_out, int out_size, void* d_ws, size_t ws_size,
                              hipStream_t stream) {
    const int*   species  = (const int*)d_in[0];
    const float* energies = (const float*)d_in[1];
    const int*   idx12    = (const int*)d_in[2];
    const float* dist     = (const float*)d_in[3];
    const float* covr     = (const float*)d_in[4];
    const float* cutoff   = (const float*)d_in[5];
    const float* qtab     = (const float*)d_in[6];
    const float* t6       = (const float*)d_in[7];
    const float* ta       = (const float*)d_in[8];
    const float* tb       = (const float*)d_in[9];
    const float* s6       = (const float*)d_in[10];
    const float* s8       = (const float*)d_in[11];
    const float* a1       = (const float*)d_in[12];
    const float* a2       = (const float*)d_in[13];

    const int natoms = in_sizes[0];        // 5000
    const int npairs = in_sizes[3];        // 2,000,000

    float* coord = (float*)d_ws;           // [natoms]
    float* accum = coord + natoms;         // [1]

    const int nzero = natoms + 1;
    zero_ws_kernel<<<(nzero + 255) / 256, 256, 0, stream>>>(coord, nzero);

    if (npairs > 0) {
        cn_kernel<<<2048, 256, 0, stream>>>(species, idx12, dist, covr,
                                            coord, npairs);
        energy_kernel<<<2048, TPB, 0, stream>>>(species, idx12, dist,
                                                cutoff, qtab, t6, ta, tb,
                                                coord, s6, s8, a1, a2,
                                                accum, npairs);
    }

    finalize_kernel<<<1, 1, 0, stream>>>(energies, accum, (float*)d_out);
}